// MultiHeadAdapter_22058952032680
// MI455X (gfx1250) — compile-verified
//
#include <hip/hip_runtime.h>

typedef float v2f __attribute__((ext_vector_type(2)));
typedef float v8f __attribute__((ext_vector_type(8)));

#define H_ 8
#define D_ 128
#define K_ 32
#define NTOK 32768          // B*S
#define CHUNK_TOK 64
#define CHUNKS 4
#define TOKGRP 128          // gridDim.x ; TOKGRP*CHUNKS*CHUNK_TOK == NTOK
#define NORM_STRIDE 132     // 132 mod 64 = 4 -> conflict-free v2f A reads
#define MID_STRIDE 36

__device__ __forceinline__ float wave_sum(float v) {
  v += __shfl_xor(v, 1, 32);
  v += __shfl_xor(v, 2, 32);
  v += __shfl_xor(v, 4, 32);
  v += __shfl_xor(v, 8, 32);
  v += __shfl_xor(v, 16, 32);
  return v;
}

__device__ __forceinline__ float gelu_exact(float v) {
  return 0.5f * v * (1.0f + erff(v * 0.70710678118654752440f));
}

// PASS1: compute y = up(gelu(down(ln(x)))) and emit per-block column sums / sumsq.
// PASS2 (PASS1=false): recompute y, apply pre-folded BN affine (G,SH) + residual.
template<bool PASS1>
__global__ __launch_bounds__(256) void adapter_pass(
    const float* __restrict__ x,
    const float* __restrict__ ln_g, const float* __restrict__ ln_b,
    const float* __restrict__ wd,  const float* __restrict__ bdn,
    const float* __restrict__ wu,  const float* __restrict__ bup,
    const float* __restrict__ gsh,   // PASS2: [f]=G=0.1*gamma*rs, [1024+f]=SH
    float* __restrict__ out,         // PASS2
    float* __restrict__ partial)     // PASS1: [(h*128+bx)*256 + {0:128 sum,128:256 sq}]
{
  __shared__ float s_norm[CHUNK_TOK * NORM_STRIDE];
  __shared__ float s_mid[CHUNK_TOK * MID_STRIDE];
  __shared__ float s_red[2 * 8 * 64];

  const int tid  = threadIdx.x;
  const int lane = tid & 31;
  const int w    = tid >> 5;      // wave 0..7
  const int hi   = lane >> 4;     // lane half selects K pair / M+8
  const int ln16 = lane & 15;
  const int h    = blockIdx.y;
  const int bx   = blockIdx.x;

  const float* __restrict__ wdh = wd + h * (D_ * K_);
  const float* __restrict__ wuh = wu + h * (K_ * D_);

  const float4 lg = *reinterpret_cast<const float4*>(ln_g + h * D_ + lane * 4);
  const float4 lb = *reinterpret_cast<const float4*>(ln_b + h * D_ + lane * 4);

  const int mtile = w >> 1;       // 0..3 : 16-token row tile
  const int cgrp  = w & 1;        // column group

  float accS[4] = {0.f, 0.f, 0.f, 0.f};
  float accQ[4] = {0.f, 0.f, 0.f, 0.f};

  for (int chunk = 0; chunk < CHUNKS; ++chunk) {
    const int tokbase = (bx * CHUNKS + chunk) * CHUNK_TOK;

    // ---- LayerNorm: one wave per token row, 8 tokens per wave ----
    for (int i = 0; i < 8; ++i) {
      const int t = w * 8 + i;
      const size_t off = (size_t)(tokbase + t) * (H_ * D_) + h * D_ + lane * 4;
      const float4 v = *reinterpret_cast<const float4*>(x + off);
      const float s = wave_sum(v.x + v.y + v.z + v.w);
      const float mean = s * (1.0f / 128.0f);
      const float d0 = v.x - mean, d1 = v.y - mean, d2 = v.z - mean, d3 = v.w - mean;
      const float q = wave_sum(d0 * d0 + d1 * d1 + d2 * d2 + d3 * d3);
      const float rs = rsqrtf(q * (1.0f / 128.0f) + 1e-5f);
      float4 nv;
      nv.x = d0 * rs * lg.x + lb.x;
      nv.y = d1 * rs * lg.y + lb.y;
      nv.z = d2 * rs * lg.z + lb.z;
      nv.w = d3 * rs * lg.w + lb.w;
      *reinterpret_cast<float4*>(&s_norm[t * NORM_STRIDE + lane * 4]) = nv;
    }
    __syncthreads();

    // ---- GEMM1: (16x128) x (128x32) -> one 16x16 tile per wave ----
    {
      const int n0 = cgrp * 16;
      v8f acc = {0.f, 0.f, 0.f, 0.f, 0.f, 0.f, 0.f, 0.f};
      for (int k0 = 0; k0 < D_; k0 += 4) {
        const v2f a = *reinterpret_cast<const v2f*>(
            &s_norm[(mtile * 16 + ln16) * NORM_STRIDE + k0 + 2 * hi]);
        v2f b;
        b[0] = wdh[(k0 + 2 * hi + 0) * K_ + n0 + ln16];
        b[1] = wdh[(k0 + 2 * hi + 1) * K_ + n0 + ln16];
        acc = __builtin_amdgcn_wmma_f32_16x16x4_f32(false, a, false, b,
                                                    (short)0, acc, false, false);
      }
      const float bias = bdn[h * K_ + n0 + ln16];
      #pragma unroll
      for (int r = 0; r < 8; ++r) {
        const float mval = gelu_exact(acc[r] + bias);
        s_mid[(mtile * 16 + r + 8 * hi) * MID_STRIDE + n0 + ln16] = mval;
      }
    }
    __syncthreads();

    // ---- GEMM2: (16x32) x (32x128) -> 4 column tiles per wave ----
    #pragma unroll
    for (int j = 0; j < 4; ++j) {
      const int n0 = cgrp * 64 + j * 16;
      v8f acc = {0.f, 0.f, 0.f, 0.f, 0.f, 0.f, 0.f, 0.f};
      for (int k0 = 0; k0 < K_; k0 += 4) {
        const v2f a = *reinterpret_cast<const v2f*>(
            &s_mid[(mtile * 16 + ln16) * MID_STRIDE + k0 + 2 * hi]);
        v2f b;
        b[0] = wuh[(k0 + 2 * hi + 0) * D_ + n0 + ln16];
        b[1] = wuh[(k0 + 2 * hi + 1) * D_ + n0 + ln16];
        acc = __builtin_amdgcn_wmma_f32_16x16x4_f32(false, a, false, b,
                                                    (short)0, acc, false, false);
      }
      const float bias = bup[h * D_ + n0 + ln16];
      if constexpr (PASS1) {
        float cs = 0.f, cq = 0.f;
        #pragma unroll
        for (int r = 0; r < 8; ++r) {
          const float yv = acc[r] + bias;
          cs += yv;
          cq += yv * yv;
        }
        cs += __shfl_xor(cs, 16, 32);   // fold the M+8 half onto the M half
        cq += __shfl_xor(cq, 16, 32);
        accS[j] += cs;
        accQ[j] += cq;
      } else {
        const float G  = gsh[h * D_ + n0 + ln16];
        const float SH = gsh[1024 + h * D_ + n0 + ln16];
        #pragma unroll
        for (int r = 0; r < 8; ++r) {
          const int tok = tokbase + mtile * 16 + r + 8 * hi;
          const size_t off = (size_t)tok * (H_ * D_) + h * D_ + n0 + ln16;
          out[off] = (acc[r] + bias) * G + SH + x[off];
        }
      }
    }
    __syncthreads();
  }

  if constexpr (PASS1) {
    if (hi == 0) {                       // lanes 0..15 hold the folded column sums
      #pragma unroll
      for (int j = 0; j < 4; ++j) {
        s_red[w * 64 + j * 16 + ln16]       = accS[j];
        s_red[512 + w * 64 + j * 16 + ln16] = accQ[j];
      }
    }
    __syncthreads();
    if (tid < 128) {                     // one thread per feature column d
      const int col = tid;
      const int grp = col >> 6;          // which cgrp owns this column
      const int cj  = col & 63;
      float s = 0.f, q = 0.f;
      for (int ww = grp; ww < 8; ww += 2) {
        s += s_red[ww * 64 + cj];
        q += s_red[512 + ww * 64 + cj];
      }
      const size_t base = (size_t)(h * TOKGRP + bx) * 256;
      partial[base + col]       = s;
      partial[base + 128 + col] = q;
    }
  }
}

// Fold per-block partials into per-feature (G, SH) with the 0.1 adapter scale pre-applied.
__global__ __launch_bounds__(256) void bn_stats(
    const float* __restrict__ partial,
    const float* __restrict__ bn_g, const float* __restrict__ bn_b,
    float* __restrict__ gsh)
{
  const int f = blockIdx.x * 256 + threadIdx.x;
  if (f >= H_ * D_) return;
  const int h = f >> 7;
  const int d = f & 127;
  float s = 0.f, q = 0.f;
  for (int tg = 0; tg < TOKGRP; ++tg) {
    const size_t base = (size_t)(h * TOKGRP + tg) * 256;
    s += partial[base + d];
    q += partial[base + 128 + d];
  }
  const float mean = s * (1.0f / (float)NTOK);
  const float var  = q * (1.0f / (float)NTOK) - mean * mean;
  const float g    = 0.1f * bn_g[f] * rsqrtf(var + 1e-5f);
  gsh[f]        = g;
  gsh[1024 + f] = 0.1f * bn_b[f] - mean * g;
}

extern "C" void kernel_launch(void* const* d_in, const int* in_sizes, int n_in,
                              void* d_out, int out_size, void* d_ws, size_t ws_size,
                              hipStream_t stream) {
  const float* x    = (const float*)d_in[0];
  const float* ln_g = (const float*)d_in[1];
  const float* ln_b = (const float*)d_in[2];
  const float* wd   = (const float*)d_in[3];
  const float* bdn  = (const float*)d_in[4];
  const float* wu   = (const float*)d_in[5];
  const float* bup  = (const float*)d_in[6];
  const float* bn_g = (const float*)d_in[7];
  const float* bn_b = (const float*)d_in[8];
  float* out = (float*)d_out;

  float* partial = (float*)d_ws;                 // 1024 blocks * 256 floats = 1 MB
  float* gsh     = partial + (size_t)H_ * TOKGRP * 256;  // 2048 floats

  dim3 grid(TOKGRP, H_);
  adapter_pass<true><<<grid, 256, 0, stream>>>(x, ln_g, ln_b, wd, bdn, wu, bup,
                                               nullptr, nullptr, partial);
  bn_stats<<<4, 256, 0, stream>>>(partial, bn_g, bn_b, gsh);
  adapter_pass<false><<<grid, 256, 0, stream>>>(x, ln_g, ln_b, wd, bdn, wu, bup,
                                                gsh, out, nullptr);
}